// Decoder_75565654606334
// MI455X (gfx1250) — compile-verified
//
#include <hip/hip_runtime.h>
#include <cmath>

// Problem constants (B, S, D, T) from the reference
#define BB 64
#define SS 512
#define DD 1024
#define TT 32

typedef float v2f __attribute__((ext_vector_type(2)));
typedef float v8f __attribute__((ext_vector_type(8)));

#if defined(__HIP_DEVICE_COMPILE__) && __has_builtin(__builtin_amdgcn_wmma_f32_16x16x4_f32)
#define USE_WMMA_F32X4 1
#else
#define USE_WMMA_F32X4 0
#endif

// ---------------------------------------------------------------------------
// Kernel 1: emissions[b,s,t] = sum_d feat[b,s,d] * W[d,t] + bias[t]
// M = B*S = 32768 rows, N = 32, K = 1024.
// One wave -> 16 rows x 32 cols (two 16x16 f32 accumulators, A reused).
// Block = 128 threads (4 waves) -> 64 rows; grid = 512 blocks.
// ---------------------------------------------------------------------------
__global__ __launch_bounds__(128) void emis_gemm_kernel(
    const float* __restrict__ feat,
    const float* __restrict__ W,
    const float* __restrict__ bias,
    float* __restrict__ emis)
{
#if USE_WMMA_F32X4
    const int lane = threadIdx.x & 31;
    const int wid  = threadIdx.x >> 5;
    const int col  = lane & 15;     // N within tile / M-row within A tile
    const int hi   = lane >> 4;     // half-wave: K pair {0,1} vs {2,3}
    const int rowBase = blockIdx.x * 64 + wid * 16;

    // A: 16x4 fp32 tile. Lane layout (ISA 7.12.2): lanes0-15 row=col K={k,k+1},
    // lanes16-31 row=col K={k+2,k+3}  ->  single b64 load at (k + 2*hi).
    const float* aptr = feat + (size_t)(rowBase + col) * DD + 2 * hi;

    v8f c0, c1;
    const float b0v = bias[col];
    const float b1v = bias[col + 16];
    #pragma unroll
    for (int r = 0; r < 8; ++r) { c0[r] = b0v; c1[r] = b1v; }

    #pragma unroll 4
    for (int k = 0; k < DD; k += 4) {
        v2f a = *(const v2f*)(aptr + k);
        const int kk = k + 2 * hi;
        // B: 4x16 fp32; lane = column, VGPR0/1 = K pair of this half-wave.
        v2f bt0, bt1;
        bt0.x = W[(kk    ) * TT + col];
        bt0.y = W[(kk + 1) * TT + col];
        bt1.x = W[(kk    ) * TT + col + 16];
        bt1.y = W[(kk + 1) * TT + col + 16];
        c0 = __builtin_amdgcn_wmma_f32_16x16x4_f32(false, a, false, bt0,
                                                   (short)0, c0, false, false);
        c1 = __builtin_amdgcn_wmma_f32_16x16x4_f32(false, a, false, bt1,
                                                   (short)0, c1, false, false);
    }

    // C/D layout: VGPR r -> row (r + 8*hi), column = lane&15 (+16 for tile 1)
    #pragma unroll
    for (int r = 0; r < 8; ++r) {
        const int orow = rowBase + r + hi * 8;
        emis[(size_t)orow * TT + col]      = c0[r];
        emis[(size_t)orow * TT + col + 16] = c1[r];
    }
#else
    // Scalar fallback (also what the host pass parses).
    const int tid = threadIdx.x;
    const int rowBase = blockIdx.x * 64;
    for (int idx = tid; idx < 64 * TT; idx += 128) {
        const int r = idx >> 5, c = idx & 31;
        const float* ap = feat + (size_t)(rowBase + r) * DD;
        float acc = bias[c];
        for (int d = 0; d < DD; ++d) acc += ap[d] * W[d * TT + c];
        emis[(size_t)(rowBase + r) * TT + c] = acc;
    }
#endif
}

// ---------------------------------------------------------------------------
// Kernel 2: CRF log-likelihood per batch element. One wave per batch;
// lane j owns state j. Forward scan in exp-domain:
//   nxt_j = e_j + m + log( sum_i exp(alpha_i - m) * exp(trans_ij) )
// with exp(trans[:, j]) pre-computed into 32 registers per lane.
// ---------------------------------------------------------------------------
__global__ __launch_bounds__(32) void crf_kernel(
    const float* __restrict__ emis,
    const int*   __restrict__ tags,
    const float* __restrict__ startT,
    const float* __restrict__ endT,
    const float* __restrict__ trans,
    float* __restrict__ llh)
{
    const int b = blockIdx.x;
    const int j = threadIdx.x;                 // target state
    const float* E  = emis + (size_t)b * SS * TT;
    const int*   tg = tags + (size_t)b * SS;

    // exp(trans[i][j]) for all source states i (column j), kept in registers.
    float texp[TT];
    #pragma unroll
    for (int i = 0; i < TT; ++i) texp[i] = __expf(trans[i * TT + j]);

    float alpha = startT[j] + E[j];            // s = 0

    for (int s = 1; s < SS; ++s) {
        const bool valid = (tg[s] != -1);
        const float e = E[s * TT + j];
        // wave max of alpha
        float m = alpha;
        #pragma unroll
        for (int off = 16; off > 0; off >>= 1)
            m = fmaxf(m, __shfl_xor(m, off, 32));
        const float p = __expf(alpha - m);
        // 32x32 matvec: acc_j = sum_i p_i * exp(trans_ij)
        float a0 = 0.f, a1 = 0.f, a2 = 0.f, a3 = 0.f;
        #pragma unroll
        for (int i = 0; i < TT; i += 4) {
            a0 = fmaf(__shfl(p, i    , 32), texp[i    ], a0);
            a1 = fmaf(__shfl(p, i + 1, 32), texp[i + 1], a1);
            a2 = fmaf(__shfl(p, i + 2, 32), texp[i + 2], a2);
            a3 = fmaf(__shfl(p, i + 3, 32), texp[i + 3], a3);
        }
        const float nxt = e + m + __logf((a0 + a1) + (a2 + a3));
        alpha = valid ? nxt : alpha;
    }

    // log partition function
    float v = alpha + endT[j];
    float m2 = v;
    #pragma unroll
    for (int off = 16; off > 0; off >>= 1)
        m2 = fmaxf(m2, __shfl_xor(m2, off, 32));
    float se = __expf(v - m2);
    #pragma unroll
    for (int off = 16; off > 0; off >>= 1)
        se += __shfl_xor(se, off, 32);
    const float logz = m2 + __logf(se);

    // gold path score: lanes stride over sequence positions
    float sc = 0.f;
    int cnt = 0;
    for (int s = j; s < SS; s += 32) {
        const int t = tg[s];
        const bool vld = (t != -1);
        const int tc = vld ? t : 0;
        cnt += vld ? 1 : 0;
        if (s == 0) {
            sc += startT[tc] + E[tc];
        } else {
            const int tp  = tg[s - 1];
            const int tpc = (tp != -1) ? tp : 0;
            const float term = trans[tpc * TT + tc] + E[s * TT + tc];
            sc += vld ? term : 0.f;
        }
    }
    #pragma unroll
    for (int off = 16; off > 0; off >>= 1) {
        sc  += __shfl_xor(sc,  off, 32);
        cnt += __shfl_xor(cnt, off, 32);
    }
    if (j == 0) {
        const int last = tg[cnt - 1];          // mask is monotone (prefix)
        sc += endT[last];
        llh[b] = sc - logz;
    }
}

// ---------------------------------------------------------------------------
// Kernel 3: loss = -mean(llh) over B = 64
// ---------------------------------------------------------------------------
__global__ __launch_bounds__(32) void loss_kernel(const float* __restrict__ llh,
                                                  float* __restrict__ out)
{
    const int j = threadIdx.x;
    float s = llh[j] + llh[j + 32];
    #pragma unroll
    for (int off = 16; off > 0; off >>= 1)
        s += __shfl_xor(s, off, 32);
    if (j == 0) out[0] = -s / (float)BB;
}

// ---------------------------------------------------------------------------
extern "C" void kernel_launch(void* const* d_in, const int* in_sizes, int n_in,
                              void* d_out, int out_size, void* d_ws, size_t ws_size,
                              hipStream_t stream)
{
    const float* feat = (const float*)d_in[0];   // [B,S,D] f32
    const int*   tags = (const int*)  d_in[1];   // [B,S]   int
    const float* W    = (const float*)d_in[2];   // [D,T]   f32
    const float* bias = (const float*)d_in[3];   // [T]
    const float* st   = (const float*)d_in[4];   // [T]
    const float* en   = (const float*)d_in[5];   // [T]
    const float* tr   = (const float*)d_in[6];   // [T,T]

    float* emis = (float*)d_ws;                      // B*S*T floats = 4 MB
    float* llh  = emis + (size_t)BB * SS * TT;       // + B floats

    emis_gemm_kernel<<<dim3((BB * SS) / 64), dim3(128), 0, stream>>>(feat, W, bias, emis);
    crf_kernel<<<dim3(BB), dim3(32), 0, stream>>>(emis, tags, st, en, tr, llh);
    loss_kernel<<<dim3(1), dim3(32), 0, stream>>>(llh, (float*)d_out);
}